// PointNetSetAbstraction_89721866814220
// MI455X (gfx1250) — compile-verified
//
#include <hip/hip_runtime.h>
#include <hip/hip_bf16.h>

typedef __attribute__((ext_vector_type(2))) float v2f;
typedef __attribute__((ext_vector_type(8))) float v8f;

#define BATCH   32
#define NPTS    32768
#define NSAMP   4096
#define FPS_BT  1024
#define FPS_CH  8    // chunks of 4 consecutive points per thread -> 32 pts/thread

// ---------------------------------------------------------------------------
// Furthest point sampling: one workgroup per batch. dist[] lives in VGPRs
// (32 floats/thread); xyz streamed from L2 (384KB/batch, L2-resident) with
// float4 (global_load_b128) loads: thread t, chunk j covers points
// j*4096 + 4*t + {0..3}; per wave each row load is 512B contiguous.
// Argmax: per-point float cmp/select, then one packed (bits(d)<<32)|idx
// 64-bit key per iter -> wave32 shuffle max -> wave0 reduces 32 leaders.
// ---------------------------------------------------------------------------
__global__ void __launch_bounds__(FPS_BT)
fps_kernel(const float* __restrict__ xyz, int* __restrict__ idx) {
    __shared__ unsigned long long red[FPS_BT / 32];
    __shared__ int s_last;

    const int b = blockIdx.x;
    const int t = threadIdx.x;
    const float* xb = xyz + (size_t)b * 3 * NPTS;

    float dist[FPS_CH * 4];
#pragma unroll
    for (int i = 0; i < FPS_CH * 4; ++i) dist[i] = 3.4e38f;

    int last = 0;
    for (int it = 0; it < NSAMP; ++it) {
        if (t == 0) idx[(size_t)b * NSAMP + it] = last;   // scan emits previous 'last'

        // broadcast load of the last-picked point (same address in all lanes)
        const float lx = xb[last];
        const float ly = xb[NPTS + last];
        const float lz = xb[2 * NPTS + last];

        float bestD = -1.f;
        int bestIdx = 0;
#pragma unroll
        for (int j = 0; j < FPS_CH; ++j) {
            const int base = j * (FPS_BT * 4) + t * 4;
            const float4 px = *(const float4*)(xb + base);
            const float4 py = *(const float4*)(xb + NPTS + base);
            const float4 pz = *(const float4*)(xb + 2 * NPTS + base);
            const float xs[4] = {px.x, px.y, px.z, px.w};
            const float ys[4] = {py.x, py.y, py.z, py.w};
            const float zs[4] = {pz.x, pz.y, pz.z, pz.w};
#pragma unroll
            for (int e = 0; e < 4; ++e) {
                const float dx = xs[e] - lx;
                const float dy = ys[e] - ly;
                const float dz = zs[e] - lz;
                const float d = fmaf(dz, dz, fmaf(dy, dy, dx * dx));
                const float dd = fminf(dist[j * 4 + e], d);
                dist[j * 4 + e] = dd;
                if (dd > bestD) { bestD = dd; bestIdx = base + e; }
            }
        }

        unsigned long long best =
            ((unsigned long long)__float_as_uint(bestD) << 32) | (unsigned)bestIdx;
        // in-wave max reduce (wave32)
#pragma unroll
        for (int off = 16; off > 0; off >>= 1) {
            const unsigned long long o = __shfl_xor(best, off, 32);
            best = (o > best) ? o : best;
        }
        if ((t & 31) == 0) red[t >> 5] = best;
        __syncthreads();
        if (t < 32) {
            unsigned long long v = red[t];
#pragma unroll
            for (int off = 16; off > 0; off >>= 1) {
                const unsigned long long o = __shfl_xor(v, off, 32);
                v = (o > v) ? o : v;
            }
            if (t == 0) s_last = (int)(v & 0xffffffffu);
        }
        __syncthreads();
        last = s_last;
    }
}

// ---------------------------------------------------------------------------
// Gather sampled points, build 6-channel feature [x,y,z, px-x,py-y,pz-z],
// pad to 8 channels (zeros) so the WMMA K-loop stays a multiple of 4.
// feat layout: [B, 8, S]
// ---------------------------------------------------------------------------
__global__ void gather_feat_kernel(const float* __restrict__ xyz,
                                   const float* __restrict__ pts,
                                   const int* __restrict__ idx,
                                   float* __restrict__ feat) {
    const int i = blockIdx.x * blockDim.x + threadIdx.x;
    if (i >= BATCH * NSAMP) return;
    const int b = i / NSAMP, s = i % NSAMP;
    const int p = idx[(size_t)b * NSAMP + s];
    const float* xb = xyz + (size_t)b * 3 * NPTS;
    const float* pb = pts + (size_t)b * 3 * NPTS;
    const float x = xb[p], y = xb[NPTS + p], z = xb[2 * NPTS + p];
    float* fb = feat + (size_t)b * 8 * NSAMP;
    fb[0 * NSAMP + s] = x;
    fb[1 * NSAMP + s] = y;
    fb[2 * NSAMP + s] = z;
    fb[3 * NSAMP + s] = pb[p] - x;
    fb[4 * NSAMP + s] = pb[NPTS + p] - y;
    fb[5 * NSAMP + s] = pb[2 * NPTS + p] - z;
    fb[6 * NSAMP + s] = 0.f;
    fb[7 * NSAMP + s] = 0.f;
}

// Pad W0 [64,6] -> [64,8] (zero cols 6,7)
__global__ void pad_w0_kernel(const float* __restrict__ W0, float* __restrict__ Wp) {
    const int i = blockIdx.x * blockDim.x + threadIdx.x;
    if (i >= 64 * 8) return;
    const int r = i >> 3, c = i & 7;
    Wp[i] = (c < 6) ? W0[r * 6 + c] : 0.f;
}

// ---------------------------------------------------------------------------
// Y[b] = W @ X[b] + bias, via V_WMMA_F32_16X16X4_F32 (full fp32 precision).
// One wave per block; each wave produces a 16(M) x 128(N) strip: the A
// (weight) fragment is loaded once per K-step and reused across 8 WMMAs.
//   A frag (16x4):  lane L -> M = L&15,  K = k + 2*(L>>4) + v   (v in {0,1})
//   B frag (4x16):  lane L -> N = L&15,  K = k + 2*(L>>4) + v
//   C/D (16x16):    VGPR v -> M = v + 8*(L>>4), N = L&15
// EXEC must be all-ones for WMMA: block = exactly one full wave, no branches.
// ---------------------------------------------------------------------------
#define GEMM_NT 8   // N-tiles per wave

__global__ void __launch_bounds__(32)
wmma_gemm_bias_kernel(const float* __restrict__ X, const float* __restrict__ W,
                      const float* __restrict__ bias, float* __restrict__ Y,
                      int Cin, int Cout) {
    const int lane = threadIdx.x;
    const int nn   = lane & 15;
    const int half = lane >> 4;
    const int n0   = blockIdx.x * (16 * GEMM_NT);
    const int m0   = blockIdx.y * 16;
    const int b    = blockIdx.z;
    const size_t Xb = (size_t)b * Cin * NSAMP;
    const size_t Yb = (size_t)b * Cout * NSAMP;

    v8f acc[GEMM_NT];
#pragma unroll
    for (int j = 0; j < GEMM_NT; ++j) acc[j] = (v8f){};

    for (int k = 0; k < Cin; k += 4) {
        const int krow = k + 2 * half;
        v2f a;
        a.x = W[(m0 + nn) * Cin + krow + 0];
        a.y = W[(m0 + nn) * Cin + krow + 1];
#pragma unroll
        for (int j = 0; j < GEMM_NT; ++j) {
            const int n = n0 + 16 * j + nn;
            v2f bm;
            bm.x = X[Xb + (size_t)(krow + 0) * NSAMP + n];
            bm.y = X[Xb + (size_t)(krow + 1) * NSAMP + n];
            acc[j] = __builtin_amdgcn_wmma_f32_16x16x4_f32(
                false, a, false, bm, (short)0, acc[j], false, false);
        }
    }

#pragma unroll
    for (int j = 0; j < GEMM_NT; ++j) {
        const int n = n0 + 16 * j + nn;
#pragma unroll
        for (int v = 0; v < 8; ++v) {
            const int m = m0 + v + 8 * half;
            Y[Yb + (size_t)m * NSAMP + n] = acc[j][v] + bias[m];
        }
    }
}

// ---------------------------------------------------------------------------
// BN stats (training mode, biased var over batch*spatial): one block per
// channel; fold into scale/shift so apply is a single FMA + ReLU.
// ---------------------------------------------------------------------------
__global__ void __launch_bounds__(256)
bn_stats_kernel(const float* __restrict__ Y, const float* __restrict__ g,
                const float* __restrict__ bt, float* __restrict__ scale,
                float* __restrict__ shift, int C) {
    __shared__ float ssum[256], ssq[256];
    const int c = blockIdx.x;
    const int t = threadIdx.x;
    float sum = 0.f, sq = 0.f;
    for (int b = 0; b < BATCH; ++b) {
        const float* row = Y + ((size_t)b * C + c) * NSAMP;
        for (int s = t; s < NSAMP; s += 256) {
            const float v = row[s];
            sum += v;
            sq += v * v;
        }
    }
    ssum[t] = sum;
    ssq[t] = sq;
    __syncthreads();
    for (int off = 128; off > 0; off >>= 1) {
        if (t < off) {
            ssum[t] += ssum[t + off];
            ssq[t] += ssq[t + off];
        }
        __syncthreads();
    }
    if (t == 0) {
        const float inv_m = 1.0f / (float)(BATCH * NSAMP);
        const float mean = ssum[0] * inv_m;
        const float var = ssq[0] * inv_m - mean * mean;
        const float inv = rsqrtf(var + 1e-5f);
        const float sc = g[c] * inv;
        scale[c] = sc;
        shift[c] = bt[c] - mean * sc;
    }
}

__global__ void bn_apply_kernel(float* __restrict__ Y, const float* __restrict__ scale,
                                const float* __restrict__ shift, int C) {
    const size_t total = (size_t)BATCH * C * NSAMP;
    const size_t i = (size_t)blockIdx.x * blockDim.x + threadIdx.x;
    if (i >= total) return;
    const int c = (int)((i / NSAMP) % C);
    const float v = Y[i] * scale[c] + shift[c];
    Y[i] = v > 0.f ? v : 0.f;
}

// ---------------------------------------------------------------------------
extern "C" void kernel_launch(void* const* d_in, const int* in_sizes, int n_in,
                              void* d_out, int out_size, void* d_ws, size_t ws_size,
                              hipStream_t stream) {
    (void)in_sizes; (void)n_in; (void)out_size; (void)ws_size;
    const float* xyz = (const float*)d_in[0];
    const float* pts = (const float*)d_in[1];
    const float* W0 = (const float*)d_in[2];
    const float* b0 = (const float*)d_in[3];
    const float* g0 = (const float*)d_in[4];
    const float* bt0 = (const float*)d_in[5];
    const float* W1 = (const float*)d_in[6];
    const float* b1 = (const float*)d_in[7];
    const float* g1 = (const float*)d_in[8];
    const float* bt1 = (const float*)d_in[9];
    const float* W2 = (const float*)d_in[10];
    const float* b2 = (const float*)d_in[11];
    const float* g2 = (const float*)d_in[12];
    const float* bt2 = (const float*)d_in[13];
    float* out = (float*)d_out;   // [B, 256, S]

    // workspace carve-up
    char* ws = (char*)d_ws;
    size_t off = 0;
    auto take = [&](size_t bytes) {
        void* p = ws + off;
        off = (off + bytes + 255) & ~(size_t)255;
        return p;
    };
    int*   idx   = (int*)  take((size_t)BATCH * NSAMP * sizeof(int));
    float* feat  = (float*)take((size_t)BATCH * 8 * NSAMP * sizeof(float));
    float* Wp    = (float*)take(64 * 8 * sizeof(float));
    float* act1  = (float*)take((size_t)BATCH * 64 * NSAMP * sizeof(float));
    float* act2  = (float*)take((size_t)BATCH * 128 * NSAMP * sizeof(float));
    float* scale = (float*)take(256 * sizeof(float));
    float* shift = (float*)take(256 * sizeof(float));

    // 1) FPS: one workgroup per batch
    fps_kernel<<<BATCH, FPS_BT, 0, stream>>>(xyz, idx);

    // 2) gather + feature build (+pad to 8 channels)
    {
        const int total = BATCH * NSAMP;
        gather_feat_kernel<<<(total + 255) / 256, 256, 0, stream>>>(xyz, pts, idx, feat);
    }
    pad_w0_kernel<<<2, 256, 0, stream>>>(W0, Wp);

    const int ntiles = NSAMP / (16 * GEMM_NT);   // 32

    // 3) layer 0: [64 x 8] @ [8 x S]
    {
        dim3 grid(ntiles, 64 / 16, BATCH);
        wmma_gemm_bias_kernel<<<grid, 32, 0, stream>>>(feat, Wp, b0, act1, 8, 64);
        bn_stats_kernel<<<64, 256, 0, stream>>>(act1, g0, bt0, scale, shift, 64);
        const size_t total = (size_t)BATCH * 64 * NSAMP;
        bn_apply_kernel<<<(unsigned)((total + 255) / 256), 256, 0, stream>>>(act1, scale, shift, 64);
    }
    // 4) layer 1: [128 x 64] @ [64 x S]
    {
        dim3 grid(ntiles, 128 / 16, BATCH);
        wmma_gemm_bias_kernel<<<grid, 32, 0, stream>>>(act1, W1, b1, act2, 64, 128);
        bn_stats_kernel<<<128, 256, 0, stream>>>(act2, g1, bt1, scale, shift, 128);
        const size_t total = (size_t)BATCH * 128 * NSAMP;
        bn_apply_kernel<<<(unsigned)((total + 255) / 256), 256, 0, stream>>>(act2, scale, shift, 128);
    }
    // 5) layer 2: [256 x 128] @ [128 x S] -> d_out, BN+ReLU in place
    {
        dim3 grid(ntiles, 256 / 16, BATCH);
        wmma_gemm_bias_kernel<<<grid, 32, 0, stream>>>(act2, W2, b2, out, 128, 256);
        bn_stats_kernel<<<256, 256, 0, stream>>>(out, g2, bt2, scale, shift, 256);
        const size_t total = (size_t)BATCH * 256 * NSAMP;
        bn_apply_kernel<<<(unsigned)((total + 255) / 256), 256, 0, stream>>>(out, scale, shift, 256);
    }
}